// ROIPooling2d_8452495639191
// MI455X (gfx1250) — compile-verified
//
#include <hip/hip_runtime.h>
#include <hip/hip_bf16.h>

typedef __attribute__((ext_vector_type(2))) float v2f;
typedef __attribute__((ext_vector_type(8))) float v8f;

#define ROI_S   7
#define FH      50
#define FW      50
#define NCH     256
#define KCH     13          // ceil(50/4) K-chunks of 4 (pad 50 -> 52)
#define SSCALE  0.0625f

// One block per ROI, 8 waves of 32 lanes, each wave owns 32 channels.
// Step 1: t[i,x] = sum_y Wh[i,y] * X[c,y,x]   (WMMA f32 16x16x4, 4 x-tiles x 13 K)
// Step 2: out[i,j] = sum_x t[i,x] * Ww[j,x]   (WMMA f32 16x16x4, 13 K)
//
// Out-of-range y/x (padding to 52 / 64) are handled by CLAMPING the load
// address to the last valid row/col instead of masking: the corresponding
// A (Wh) / B (Ww) weight entries are structurally zero there, so the
// clamped (finite) values contribute exactly 0. This keeps every global
// load unconditional -> no exec-mask branches in the hot loop.
__global__ __launch_bounds__(256) void roipool_wmma_kernel(
    const float* __restrict__ x,        // [4,256,50,50]
    const float* __restrict__ rois,     // [128,4] = y0,x0,y1,x1 (pre-scale)
    const int*   __restrict__ roi_idx,  // [128]
    float* __restrict__ out)            // [128,256,7,7]
{
    __shared__ float t_lds[8][16 * 64];   // per-wave 16x64 t tile (32 KB total)

    const int r    = blockIdx.x;
    const int lane = threadIdx.x & 31;
    const int wave = threadIdx.x >> 5;
    const int lmod = lane & 15;           // M/N index within tile
    const int lhi  = lane >> 4;           // K-half selector

    // ---- ROI geometry (uniform across block) ----
    const int y0 = (int)floorf(rois[r * 4 + 0] * SSCALE);
    const int x0 = (int)floorf(rois[r * 4 + 1] * SSCALE);
    const int y1 = (int)floorf(rois[r * 4 + 2] * SSCALE);
    const int x1 = (int)floorf(rois[r * 4 + 3] * SSCALE);
    const int b  = roi_idx[r];
    const int ny = y1 - y0 + 1;
    const int nx = x1 - x0 + 1;

    // ---- Build weight fragments in registers ----
    // A-matrix (Wh) fragment k: lane l, elem v -> Wh[i = lmod, y = 4k + 2*lhi + v]
    // B-matrix (Ww^T) fragment k: lane l, elem v -> Ww[j = lmod, x = 4k + 2*lhi + v]
    v2f Awh[KCH];
    v2f Bww[KCH];
    #pragma unroll
    for (int k = 0; k < KCH; ++k) {
        #pragma unroll
        for (int v = 0; v < 2; ++v) {
            const int q = 4 * k + 2 * lhi + v;   // feature coordinate (y or x)
            float wh = 0.0f, ww = 0.0f;
            if (lmod < ROI_S) {
                // Wh: adaptive-pool bin lmod along y
                const int posy = q - y0;
                const int sy = (lmod * ny) / ROI_S;
                const int ey = ((lmod + 1) * ny + ROI_S - 1) / ROI_S;
                if (posy >= sy && posy < ey) wh = 1.0f / (float)(ey - sy);
                // Ww: adaptive-pool bin lmod along x
                const int posx = q - x0;
                const int sx = (lmod * nx) / ROI_S;
                const int ex = ((lmod + 1) * nx + ROI_S - 1) / ROI_S;
                if (posx >= sx && posx < ex) ww = 1.0f / (float)(ex - sx);
            }
            Awh[k][v] = wh;
            Bww[k][v] = ww;
        }
    }

    float* tw = t_lds[wave];
    const long img_base = (long)b * NCH * FH * FW;

    for (int cc = 0; cc < 32; ++cc) {
        const int c = wave * 32 + cc;
        const float* xc = x + img_base + (long)c * FH * FW;

        // ---------- Step 1: t = Wh @ X_c over 4 x-tiles ----------
        #pragma unroll
        for (int xt = 0; xt < 4; ++xt) {
            const int xx    = xt * 16 + lmod;                  // B N-index -> x col
            const int xsafe = (xt == 3) ? min(xx, FW - 1) : xx; // clamp (weight==0 past 49)
            v8f D = {};
            #pragma unroll
            for (int k = 0; k < KCH; ++k) {
                const int yb = 4 * k + 2 * lhi;                // this lane's K base (y row)
                int yc0 = yb, yc1 = yb + 1;
                if (k == KCH - 1) {                            // only chunk 12 can exceed 49
                    yc0 = min(yc0, FH - 1);
                    yc1 = min(yc1, FH - 1);
                }
                v2f Bx;
                Bx[0] = xc[yc0 * FW + xsafe];
                Bx[1] = xc[yc1 * FW + xsafe];
                D = __builtin_amdgcn_wmma_f32_16x16x4_f32(
                        /*neg_a=*/false, Awh[k], /*neg_b=*/false, Bx,
                        /*c_mod=*/(short)0, D, /*reuse_a=*/false, /*reuse_b=*/false);
            }
            // Store D to LDS t: VGPR v, lane l -> t[(v + 8*lhi)*64 + xt*16 + lmod]
            const int tb = (8 * lhi) * 64 + xt * 16 + lmod;
            #pragma unroll
            for (int v = 0; v < 8; ++v)
                tw[tb + v * 64] = D[v];
        }

        // ---------- Step 2: out = t @ Ww^T ----------
        v8f D2 = {};
        #pragma unroll
        for (int k = 0; k < KCH; ++k) {
            // A2 fragment: lane l, elem v -> t[i = lmod, x = 4k + 2*lhi + v]
            // (two per-lane K values contiguous -> single 8-byte LDS load)
            const float* tp = tw + lmod * 64 + 4 * k + 2 * lhi;
            v2f A2; A2[0] = tp[0]; A2[1] = tp[1];
            D2 = __builtin_amdgcn_wmma_f32_16x16x4_f32(
                    false, A2, false, Bww[k], (short)0, D2, false, false);
        }

        // D2: VGPR v, lane l -> (i = v + 8*lhi, j = lmod); valid i,j < 7
        if (lhi == 0 && lmod < ROI_S) {
            float* op = out + (((long)r * NCH + c) * ROI_S) * ROI_S + lmod;
            #pragma unroll
            for (int v = 0; v < ROI_S; ++v)
                op[v * ROI_S] = D2[v];
        }
    }
}

extern "C" void kernel_launch(void* const* d_in, const int* in_sizes, int n_in,
                              void* d_out, int out_size, void* d_ws, size_t ws_size,
                              hipStream_t stream) {
    const float* x       = (const float*)d_in[0];
    const float* rois    = (const float*)d_in[1];
    const int*   roi_idx = (const int*)d_in[2];
    float*       out     = (float*)d_out;

    const int R = in_sizes[2];   // 128 ROIs
    roipool_wmma_kernel<<<R, 256, 0, stream>>>(x, rois, roi_idx, out);
}